// SignedAttention_71614284693815
// MI455X (gfx1250) — compile-verified
//
#include <hip/hip_runtime.h>
#include <math.h>

typedef __attribute__((ext_vector_type(2))) float v2f;
typedef __attribute__((ext_vector_type(4))) float v4f;
typedef __attribute__((ext_vector_type(8))) float v8f;

// Problem constants (from reference)
constexpr int LSEQ    = 1024;
constexpr int DHEAD   = 64;
constexpr int NPATCH  = 128;
constexpr int CH      = LSEQ / NPATCH;   // 8 channels per patch
constexpr int WIN     = 10;              // attend iff 0 < kp - qp < WIN
constexpr int QROWS   = 16;              // query rows per block = 2 patches
constexpr int KSPAN   = 80;              // 10 key patches * 8 channels
constexpr int KTILES  = KSPAN / 16;      // 5 score tiles of 16 keys
constexpr int NTILES  = DHEAD / 16;      // 4 output tiles of 16 dims

__global__ __launch_bounds__(32)
void signed_attn_wmma(const float* __restrict__ Q,
                      const float* __restrict__ K,
                      const float* __restrict__ V,
                      const float* __restrict__ log_scale,
                      float* __restrict__ O)
{
    __shared__ __align__(16) float Ssm[QROWS][KSPAN];  // raw scores
    __shared__ __align__(16) float Asm[QROWS][KSPAN];  // signed-softmax weights
    __shared__ __align__(16) float Vsm[KSPAN][DHEAD];  // staged V tile

    const int lane = threadIdx.x;        // 0..31 (wave32)
    const int row  = lane & 15;          // fragment row/col within half-wave
    const int khalf = (lane >> 4) * 2;   // K-dim pair selector for f32 WMMA frags

    const int blk  = blockIdx.x;
    const int qblk = blk & 63;           // 64 query blocks per (b,h)
    const int bh   = blk >> 6;
    const int q0   = qblk * QROWS;       // first query row
    const int p0   = qblk * 2;           // first query patch
    const int k0   = (p0 + 1) * CH;      // first key position of the 80-wide span

    const float* Qb = Q + (size_t)bh * LSEQ * DHEAD;
    const float* Kb = K + (size_t)bh * LSEQ * DHEAD;
    const float* Vb = V + (size_t)bh * LSEQ * DHEAD;
    float*       Ob = O + (size_t)bh * LSEQ * DHEAD;

    const float sc = fminf(fmaxf(__expf(log_scale[0]), 1.0f), 30.0f) * 0.125f; // /sqrt(64)

    // ---------------- Phase 1: S = Q * K^T (5 tiles of 16x16, fp32 WMMA, K=4/step)
    v8f acc[KTILES];
    #pragma unroll
    for (int t = 0; t < KTILES; ++t) acc[t] = (v8f)0.0f;

    #pragma unroll
    for (int ks = 0; ks < DHEAD / 4; ++ks) {
        const int d0 = ks * 4 + khalf;
        // A fragment: lane holds Q[q0+row][d0], Q[q0+row][d0+1] (contiguous)
        v2f a = *(const v2f*)(Qb + (size_t)(q0 + row) * DHEAD + d0);
        #pragma unroll
        for (int t = 0; t < KTILES; ++t) {
            int key = k0 + t * 16 + row;
            if (key > LSEQ - 1) key = LSEQ - 1;      // clamp; masked later
            v2f b = *(const v2f*)(Kb + (size_t)key * DHEAD + d0);
            acc[t] = __builtin_amdgcn_wmma_f32_16x16x4_f32(
                false, a, false, b, (short)0, acc[t], false, false);
        }
    }

    // Spill D-fragments (row-across-lanes) to LDS for the row-wise softmax.
    #pragma unroll
    for (int t = 0; t < KTILES; ++t)
        #pragma unroll
        for (int r = 0; r < 8; ++r)
            Ssm[r + 8 * (lane >> 4)][(lane & 15) + 16 * t] = acc[t][r];
    __syncthreads();

    // ---------------- Phase 2: signed double softmax per query row (lanes 0..15)
    if (lane < 16) {
        const int r  = lane;
        const int qp = p0 + (r >> 3);                // this row's query patch
        float mp = -3.4e38f, mn = -3.4e38f;
        int any = 0;
        #pragma unroll 8
        for (int c = 0; c < KSPAN; ++c) {
            const int kp = p0 + 1 + (c >> 3);
            const int d  = kp - qp;
            if (d >= 1 && d < WIN && kp < NPATCH) {
                const float s = sc * Ssm[r][c];
                mp = fmaxf(mp, s);
                mn = fmaxf(mn, -s);
                any = 1;
            }
        }
        if (!any) {
            // fully masked row: softmax(pos) - softmax(neg) == 0 exactly
            for (int c = 0; c < KSPAN; ++c) Asm[r][c] = 0.0f;
        } else {
            float zp = 0.0f, zn = 0.0f;
            #pragma unroll 8
            for (int c = 0; c < KSPAN; ++c) {
                const int kp = p0 + 1 + (c >> 3);
                const int d  = kp - qp;
                if (d >= 1 && d < WIN && kp < NPATCH) {
                    const float s = sc * Ssm[r][c];
                    zp += __expf(s - mp);
                    zn += __expf(-s - mn);
                }
            }
            const float izp = 1.0f / zp, izn = 1.0f / zn;
            #pragma unroll 8
            for (int c = 0; c < KSPAN; ++c) {
                const int kp = p0 + 1 + (c >> 3);
                const int d  = kp - qp;
                float a = 0.0f;
                if (d >= 1 && d < WIN && kp < NPATCH) {
                    const float s = sc * Ssm[r][c];
                    a = __expf(s - mp) * izp - __expf(-s - mn) * izn;
                }
                Asm[r][c] = a;
            }
        }
    }

    // ---------------- Phase 3: stage V[k0 : k0+80][0:64] into LDS (all 32 lanes)
    {
        float4* vs = (float4*)&Vsm[0][0];
        #pragma unroll
        for (int i = 0; i < (KSPAN * DHEAD) / (32 * 4); ++i) {   // 40 float4 per lane
            const int f   = lane + i * 32;       // float4 index
            const int kr  = f >> 4;              // row within span (64 floats/row)
            const int col = (f & 15) * 4;
            int key = k0 + kr;
            if (key > LSEQ - 1) key = LSEQ - 1;  // A is 0 there; value irrelevant
            vs[f] = *(const float4*)(Vb + (size_t)key * DHEAD + col);
        }
    }
    __syncthreads();

    // ---------------- Phase 4: O = A * V  (4 tiles of 16x16, K=80)
    #pragma unroll
    for (int nt = 0; nt < NTILES; ++nt) {
        v8f oacc = (v8f)0.0f;
        #pragma unroll
        for (int ks = 0; ks < KSPAN / 4; ++ks) {
            const int kb = ks * 4 + khalf;
            // A fragment: lane holds Asm[row][kb], Asm[row][kb+1] (contiguous, 8B aligned)
            v2f a = *(const v2f*)&Asm[row][kb];
            // B fragment: lane holds V[kb][d], V[kb+1][d] for column d
            const int d = nt * 16 + row;
            v2f b;
            b.x = Vsm[kb][d];
            b.y = Vsm[kb + 1][d];
            oacc = __builtin_amdgcn_wmma_f32_16x16x4_f32(
                false, a, false, b, (short)0, oacc, false, false);
        }
        // Write D fragment: VGPR r -> row (r + 8*(lane/16)), col = lane%16
        #pragma unroll
        for (int r = 0; r < 8; ++r) {
            const int m = r + 8 * (lane >> 4);
            Ob[(size_t)(q0 + m) * DHEAD + nt * 16 + (lane & 15)] = oacc[r];
        }
    }
}

extern "C" void kernel_launch(void* const* d_in, const int* in_sizes, int n_in,
                              void* d_out, int out_size, void* d_ws, size_t ws_size,
                              hipStream_t stream) {
    const float* Q  = (const float*)d_in[0];
    const float* K  = (const float*)d_in[1];
    const float* V  = (const float*)d_in[2];
    const float* ls = (const float*)d_in[3];
    float* O = (float*)d_out;

    const int BH = in_sizes[0] / (LSEQ * DHEAD);     // B*H = 32
    dim3 grid(BH * (LSEQ / QROWS));                  // 2048 one-wave workgroups
    signed_attn_wmma<<<grid, 32, 0, stream>>>(Q, K, V, ls, O);
}